// SparsEmbed_64544768524610
// MI455X (gfx1250) — compile-verified
//
#include <hip/hip_runtime.h>
#include <hip/hip_bf16.h>
#include <math.h>

#define B_ 8
#define S_ 256
#define V_ 30522
#define H_ 768
#define E_ 256
#define K_ 256

typedef __attribute__((ext_vector_type(2))) float v2f;
typedef __attribute__((ext_vector_type(8))) float v8f;

// ---------------------------------------------------------------------------
// 1) max over seq of log1p(relu(logits)) -> maxpool[B,V]
//    (log1p is monotonic, so take max of relu first, log1p once)
__global__ void se_maxpool_kernel(const float* __restrict__ logits,
                                  float* __restrict__ maxpool) {
    int idx = blockIdx.x * blockDim.x + threadIdx.x;
    if (idx >= B_ * V_) return;
    int b = idx / V_;
    int v = idx - b * V_;
    const float* p = logits + (size_t)b * S_ * V_ + v;
    float m = 0.0f;
    #pragma unroll 4
    for (int s = 0; s < S_; ++s) {
        float x = p[(size_t)s * V_];
        m = fmaxf(m, fmaxf(x, 0.0f));
    }
    maxpool[idx] = log1pf(m);
}

// ---------------------------------------------------------------------------
// 2) zero-fill a float region (sparse activations output)
__global__ void se_zero_kernel(float* __restrict__ p, int n) {
    int i = blockIdx.x * blockDim.x + threadIdx.x;
    if (i < n) p[i] = 0.0f;
}

// ---------------------------------------------------------------------------
// 3) per-batch top-k via 4-pass MSD radix select on float bit patterns
//    (all inputs >= 0 so uint bit order == value order), then rank-sort
//    k=256 candidates (value desc, index asc) to match jax.lax.top_k.
__global__ void se_topk_kernel(const float* __restrict__ maxpool,
                               float* __restrict__ act_val,
                               int* __restrict__ act_idx,
                               float* __restrict__ out_sparse,
                               float* __restrict__ out_act_f) {
    const int b = blockIdx.x;
    const int tid = threadIdx.x;  // blockDim == 256
    const float* vals = maxpool + (size_t)b * V_;

    __shared__ unsigned s_cnt[256];
    __shared__ unsigned s_prefix;
    __shared__ unsigned s_mask;
    __shared__ int s_need;
    __shared__ unsigned s_gt, s_eq;
    __shared__ float sh_val[K_];
    __shared__ int sh_idx[K_];

    if (tid == 0) { s_prefix = 0u; s_mask = 0u; s_need = K_; s_gt = 0u; s_eq = 0u; }
    __syncthreads();

    for (int pass = 3; pass >= 0; --pass) {
        s_cnt[tid] = 0u;
        __syncthreads();
        const unsigned mask = s_mask, prefix = s_prefix;
        const int shift = pass * 8;
        for (int i = tid; i < V_; i += 256) {
            unsigned key = __float_as_uint(vals[i]);
            if ((key & mask) == prefix)
                atomicAdd(&s_cnt[(key >> shift) & 0xFFu], 1u);
        }
        __syncthreads();
        if (tid == 0) {
            int need = s_need;
            unsigned cum = 0u;
            int chosen = 0;
            for (int bin = 255; bin >= 0; --bin) {
                unsigned c = s_cnt[bin];
                if ((int)(cum + c) >= need) { chosen = bin; break; }
                cum += c;
            }
            s_need = need - (int)cum;
            s_prefix = s_prefix | ((unsigned)chosen << shift);
            s_mask = s_mask | (0xFFu << shift);
        }
        __syncthreads();
    }

    const unsigned thr = s_prefix;   // exact key of k-th largest
    const int need_eq = s_need;      // how many ==thr elements complete the set

    for (int i = tid; i < V_; i += 256) {
        unsigned key = __float_as_uint(vals[i]);
        if (key > thr) {
            unsigned p = atomicAdd(&s_gt, 1u);
            if (p < K_) { sh_val[p] = vals[i]; sh_idx[p] = i; }
        }
    }
    __syncthreads();
    const unsigned base = s_gt;      // == K_ - need_eq by construction
    for (int i = tid; i < V_; i += 256) {
        unsigned key = __float_as_uint(vals[i]);
        if (key == thr) {
            unsigned p = atomicAdd(&s_eq, 1u);
            if ((int)p < need_eq && base + p < K_) {
                sh_val[base + p] = vals[i];
                sh_idx[base + p] = i;
            }
        }
    }
    __syncthreads();

    // rank sort: descending value, ascending index (deterministic)
    float v = sh_val[tid];
    int ix = sh_idx[tid];
    int rank = 0;
    #pragma unroll 8
    for (int j = 0; j < K_; ++j) {
        float vj = sh_val[j];
        int ij = sh_idx[j];
        rank += (vj > v) || (vj == v && ij < ix);
    }
    act_val[b * K_ + rank] = v;
    act_idx[b * K_ + rank] = ix;
    out_act_f[b * K_ + rank] = (float)ix;
    out_sparse[(size_t)b * V_ + ix] = v;
}

// ---------------------------------------------------------------------------
// 4) attention[b,j,s] = softmax_s(logits[b,s,act[b,j]])  -- one wave32 per (b,j)
__global__ void se_softmax_kernel(const float* __restrict__ logits,
                                  const int* __restrict__ act_idx,
                                  float* __restrict__ att) {
    const int gw = blockIdx.x * (blockDim.x >> 5) + (threadIdx.x >> 5);
    const int lane = threadIdx.x & 31;
    const int b = gw / K_;
    const int j = gw - b * K_;
    const int col = act_idx[b * K_ + j];
    const float* base = logits + (size_t)b * S_ * V_ + col;

    float x[8];
    float m = -INFINITY;
    #pragma unroll
    for (int r = 0; r < 8; ++r) {
        x[r] = base[(size_t)(lane + 32 * r) * V_];
        m = fmaxf(m, x[r]);
    }
    #pragma unroll
    for (int off = 16; off > 0; off >>= 1) m = fmaxf(m, __shfl_xor(m, off, 32));
    float s = 0.0f;
    #pragma unroll
    for (int r = 0; r < 8; ++r) { x[r] = __expf(x[r] - m); s += x[r]; }
    #pragma unroll
    for (int off = 16; off > 0; off >>= 1) s += __shfl_xor(s, off, 32);
    const float inv = 1.0f / s;
    float* dst = att + (size_t)(b * K_ + j) * S_;
    #pragma unroll
    for (int r = 0; r < 8; ++r) dst[lane + 32 * r] = x[r] * inv;
}

// ---------------------------------------------------------------------------
// 5) pooled[b,k,h] = sum_s attention[b,k,s] * embeddings[b,s,h]
//    one wave per 16x16 output tile, v_wmma_f32_16x16x4_f32 over S in steps of 4
__global__ void se_gemm1_kernel(const float* __restrict__ att,
                                const float* __restrict__ emb,
                                float* __restrict__ pooled) {
    const int gw = blockIdx.x * (blockDim.x >> 5) + (threadIdx.x >> 5);
    const int lane = threadIdx.x & 31;
    const int NHT = H_ / 16;            // 48 h-tiles
    const int ht = gw % NHT;
    const int rest = gw / NHT;
    const int kt = rest % (K_ / 16);
    const int b = rest / (K_ / 16);

    const int half = lane >> 4;         // 0: lanes 0-15 (K=0,1)  1: lanes 16-31 (K=2,3)
    const int l15 = lane & 15;
    const int koff = half * 2;

    const float* arow = att + (size_t)(b * K_ + kt * 16 + l15) * S_;
    const float* bcol = emb + (size_t)b * S_ * H_ + (size_t)(ht * 16 + l15);

    v8f acc = {};
    for (int s0 = 0; s0 < S_; s0 += 4) {
        v2f a = *(const v2f*)(arow + s0 + koff);                 // A[m][koff..koff+1]
        v2f bb;
        bb.x = bcol[(size_t)(s0 + koff) * H_];                   // B[koff][n]
        bb.y = bcol[(size_t)(s0 + koff + 1) * H_];               // B[koff+1][n]
        acc = __builtin_amdgcn_wmma_f32_16x16x4_f32(
            false, a, false, bb, (short)0, acc, false, false);
    }
    #pragma unroll
    for (int r = 0; r < 8; ++r) {
        pooled[(size_t)(b * K_ + kt * 16 + r + half * 8) * H_ + ht * 16 + l15] = acc[r];
    }
}

// ---------------------------------------------------------------------------
// 6) out[b,k,e] = relu(sum_h pooled[b,k,h] * W[e,h] + bias[e])
__global__ void se_gemm2_kernel(const float* __restrict__ pooled,
                                const float* __restrict__ Wm,
                                const float* __restrict__ bias,
                                float* __restrict__ out) {
    const int gw = blockIdx.x * (blockDim.x >> 5) + (threadIdx.x >> 5);
    const int lane = threadIdx.x & 31;
    const int NET = E_ / 16;            // 16 e-tiles
    const int et = gw % NET;
    const int rest = gw / NET;
    const int kt = rest % (K_ / 16);
    const int b = rest / (K_ / 16);

    const int half = lane >> 4;
    const int l15 = lane & 15;
    const int koff = half * 2;

    const float* arow = pooled + (size_t)(b * K_ + kt * 16 + l15) * H_;
    const float* wrow = Wm + (size_t)(et * 16 + l15) * H_;      // W is [E,H]; B[kk][n]=W[n][kk]

    v8f acc = {};
    for (int h0 = 0; h0 < H_; h0 += 4) {
        v2f a = *(const v2f*)(arow + h0 + koff);
        v2f bb = *(const v2f*)(wrow + h0 + koff);               // consecutive h -> float2
        acc = __builtin_amdgcn_wmma_f32_16x16x4_f32(
            false, a, false, bb, (short)0, acc, false, false);
    }
    const float bn = bias[et * 16 + l15];
    #pragma unroll
    for (int r = 0; r < 8; ++r) {
        out[(size_t)(b * K_ + kt * 16 + r + half * 8) * E_ + et * 16 + l15] =
            fmaxf(acc[r] + bn, 0.0f);
    }
}

// ---------------------------------------------------------------------------
extern "C" void kernel_launch(void* const* d_in, const int* in_sizes, int n_in,
                              void* d_out, int out_size, void* d_ws, size_t ws_size,
                              hipStream_t stream) {
    const float* logits = (const float*)d_in[0];   // [B,S,V]
    const float* emb    = (const float*)d_in[1];   // [B,S,H]
    const float* Wm     = (const float*)d_in[2];   // [E,H]
    const float* bias   = (const float*)d_in[3];   // [E]
    // d_in[4] is k (==256), baked into K_

    // output layout: out [B,K,E] | sparse [B,V] | activations [B,K] (as float)
    float* out_main   = (float*)d_out;
    float* out_sparse = out_main + (size_t)B_ * K_ * E_;
    float* out_act    = out_sparse + (size_t)B_ * V_;

    // workspace layout (floats): maxpool [B,V] | actval [B,K] | actidx [B,K] | att [B,K,S] | pooled [B,K,H]
    float* ws_maxpool = (float*)d_ws;
    float* ws_actval  = ws_maxpool + (size_t)B_ * V_;
    int*   ws_actidx  = (int*)(ws_actval + (size_t)B_ * K_);
    float* ws_att     = (float*)(ws_actidx + (size_t)B_ * K_);
    float* ws_pooled  = ws_att + (size_t)B_ * K_ * S_;

    // 1) max-pool over seq (reads the 250MB logits tensor once, coalesced)
    {
        int n = B_ * V_;
        se_maxpool_kernel<<<(n + 255) / 256, 256, 0, stream>>>(logits, ws_maxpool);
    }
    // 2) zero sparse-activation output region
    {
        int n = B_ * V_;
        se_zero_kernel<<<(n + 255) / 256, 256, 0, stream>>>(out_sparse, n);
    }
    // 3) per-batch radix-select top-k + rank sort + scatter
    se_topk_kernel<<<B_, 256, 0, stream>>>(ws_maxpool, ws_actval, ws_actidx,
                                           out_sparse, out_act);
    // 4) gather + softmax -> attention [B,K,S], one wave per (b,j)
    {
        int waves = B_ * K_;                    // 2048
        se_softmax_kernel<<<waves / 4, 128, 0, stream>>>(logits, ws_actidx, ws_att);
    }
    // 5) pooled = attention @ embeddings (WMMA f32)
    {
        int waves = B_ * (K_ / 16) * (H_ / 16); // 6144 tiles
        se_gemm1_kernel<<<waves / 4, 128, 0, stream>>>(ws_att, emb, ws_pooled);
    }
    // 6) out = relu(pooled @ W^T + b) (WMMA f32)
    {
        int waves = B_ * (K_ / 16) * (E_ / 16); // 2048 tiles
        se_gemm2_kernel<<<waves / 4, 128, 0, stream>>>(ws_pooled, Wm, bias, out_main);
    }
}